// ATSSAssigner_58875411693998
// MI455X (gfx1250) — compile-verified
//
#include <hip/hip_runtime.h>

// ATSS assigner for MI455X (gfx1250, wave32).
// Compute is trivial (~1 GFLOP); traffic is output-dominated (91 MB), so the
// design keeps the (B,N,L) tensors virtual, represents positives sparsely
// (<=27 per gt), and makes the 86 MB one-hot score write coalesced 128-bit
// stores. Squared distances are computed with V_WMMA_F32_16X16X4_F32 via
// the rank-4 expansion d^2 = |g|^2 + |a|^2 - 2 g.a (exact in f32; f16/bf16
// WMMA would overflow since coords reach 1280 -> d^2 ~ 3.3e6).

#define B_SZ        8
#define N_GTS       128
#define L_ANCH      33600
#define NUM_CLASSES 80

typedef __attribute__((ext_vector_type(2))) float v2f;
typedef __attribute__((ext_vector_type(8))) float v8f;
typedef unsigned long long u64;

// ---------------------------------------------------------------- K0: init
__global__ __launch_bounds__(256) void k0_init(int* __restrict__ cnt,
                                               int* __restrict__ firstg, int n) {
  int i = blockIdx.x * 256 + threadIdx.x;
  if (i < n) { cnt[i] = 0; firstg[i] = 0x7fffffff; }
}

// ------------------------------------------- K1: WMMA distances + top-9
// One wave (32 lanes) per (16-gt tile, level). Lane l handles gt (tile*16 + l%16);
// low lanes cover anchor rows M=0..7 of each 16-anchor tile, high lanes M=8..15.
// A (16 anchors x 4) = [acx, acy, |ac|^2, 1]; B (4 x 16 gts) = [-2gcx, -2gcy, 1, |gc|^2].
// D[m,n] = |ac|^2 + |gc|^2 - 2 ac.gc = squared distance (monotone vs reference's sqrt).
// The anchor load for tile it+1 is issued before processing tile it so the
// WMMA + insertion VALU chain hides the load latency (only ~1 wave/SIMD here).
__global__ __launch_bounds__(32) void k1_topk(const float* __restrict__ anchors,
                                              const float* __restrict__ gt,
                                              int* __restrict__ topk_out) {
  const int lvOff[3]   = {0, 25600, 32000};
  const int lvTiles[3] = {1600, 400, 100};   // level anchors / 16

  int bid   = blockIdx.x;
  int level = bid % 3;
  int tile  = bid / 3;            // 0..63 over B*N/16
  int b     = tile >> 3;
  int gbase = (tile & 7) << 4;
  int lane  = threadIdx.x;
  int lm    = lane & 15;
  int hi    = lane >> 4;          // 0: K0/K1 + rows 0..7, 1: K2/K3 + rows 8..15
  int g     = gbase + lm;

  const float* gp = gt + ((size_t)(b * N_GTS + g) << 2);
  float gx1 = gp[0], gy1 = gp[1], gx2 = gp[2], gy2 = gp[3];
  float gcx = (gx1 + gx2) * 0.5f, gcy = (gy1 + gy2) * 0.5f;
  float gnorm = gcx * gcx + gcy * gcy;
  v2f Bm;                                    // loop-invariant B operand
  Bm[0] = hi ? 1.0f  : (-2.0f * gcx);        // K0 | K2
  Bm[1] = hi ? gnorm : (-2.0f * gcy);        // K1 | K3

  u64 t[9];                                  // sorted ascending (dist2_bits<<32)|idx
  #pragma unroll
  for (int k = 0; k < 9; ++k) t[k] = ~0ull;

  const int off = lvOff[level];
  const int ntiles = lvTiles[level];

  // software pipeline: av holds tile `it`, nx is being fetched for `it+1`
  float4 av = *(const float4*)(anchors + ((size_t)(off + lm) << 2));
  for (int it = 0; it < ntiles; ++it) {
    int nit = (it + 1 < ntiles) ? it + 1 : it;          // uniform, branchless
    const float4 nx =
        *(const float4*)(anchors + ((size_t)(off + (nit << 4) + lm) << 2));
    int pit = (it + 8 < ntiles) ? it + 8 : it;          // warm L2 ahead
    __builtin_prefetch(anchors + ((size_t)(off + (pit << 4) + lm) << 2), 0, 1);

    int abase = off + (it << 4);
    float acx = (av.x + av.z) * 0.5f;
    float acy = (av.y + av.w) * 0.5f;
    float anorm = acx * acx + acy * acy;
    v2f Am;
    Am[0] = hi ? anorm : acx;                // K0 | K2
    Am[1] = hi ? 1.0f  : acy;                // K1 | K3
    v8f c = {0.f, 0.f, 0.f, 0.f, 0.f, 0.f, 0.f, 0.f};
    c = __builtin_amdgcn_wmma_f32_16x16x4_f32(false, Am, false, Bm,
                                              (short)0, c, false, false);
    int mbase = abase + (hi << 3);
    #pragma unroll
    for (int v = 0; v < 8; ++v) {
      float d2 = c[v];
      d2 = d2 < 0.f ? 0.f : d2;              // rounding can produce tiny negatives
      u64 key = ((u64)__float_as_uint(d2) << 32) | (unsigned)(mbase + v);
      if (key < t[8]) {                      // insertion keeps list sorted; ties -> lower idx
        t[8] = key;
        #pragma unroll
        for (int k = 8; k > 0; --k)
          if (t[k] < t[k - 1]) { u64 tmp = t[k]; t[k] = t[k - 1]; t[k - 1] = tmp; }
      }
    }
    av = nx;
  }

  // merge partner lanes (rows 0..7 vs 8..15 of the same gt) through LDS
  __shared__ u64 lds[32 * 9];
  #pragma unroll
  for (int k = 0; k < 9; ++k) lds[lane * 9 + k] = t[k];
  __syncthreads();
  if (hi == 0) {
    const u64* mine = &lds[lane * 9];
    const u64* othr = &lds[(lane + 16) * 9];
    int* dst = topk_out + ((size_t)(b * N_GTS + g) * 3 + level) * 9;
    int i = 0, j = 0;
    #pragma unroll
    for (int k = 0; k < 9; ++k) {
      u64 a = mine[i], o = othr[j];
      u64 r;
      if (a <= o) { r = a; ++i; } else { r = o; ++j; }
      dst[k] = (int)(r & 0xffffffffu);
    }
  }
}

// ------------------------------- K2: threshold (mean + std ddof=1) + scatter
__global__ __launch_bounds__(128) void k2_thresh(const float* __restrict__ anchors,
                                                 const float* __restrict__ gt,
                                                 const float* __restrict__ pad,
                                                 const int* __restrict__ topk,
                                                 int* __restrict__ cnt,
                                                 int* __restrict__ firstg) {
  int t = blockIdx.x * 128 + threadIdx.x;
  if (t >= B_SZ * N_GTS) return;
  int b = t >> 7;
  int g = t & 127;
  if (pad[t] <= 0.f) return;                 // padded gts yield no positives

  const float* gp = gt + ((size_t)t << 2);
  float gx1 = gp[0], gy1 = gp[1], gx2 = gp[2], gy2 = gp[3];
  float garea = (gx2 - gx1) * (gy2 - gy1);

  const int* ti = topk + (size_t)t * 27;
  float iou[27];
  int   idx[27];
  float sum = 0.f;
  #pragma unroll
  for (int k = 0; k < 27; ++k) {
    int a = ti[k];
    idx[k] = a;
    const float4 av = *(const float4*)(anchors + ((size_t)a << 2));
    float lx = fmaxf(gx1, av.x), ly = fmaxf(gy1, av.y);
    float rx = fminf(gx2, av.z), ry = fminf(gy2, av.w);
    float wx = rx - lx; wx = wx < 0.f ? 0.f : wx;
    float wy = ry - ly; wy = wy < 0.f ? 0.f : wy;
    float inter = wx * wy;
    float aarea = (av.z - av.x) * (av.w - av.y);
    float v = inter / (garea + aarea - inter + 1e-9f);
    iou[k] = v;
    sum += v;
  }
  float mean = sum * (1.0f / 27.0f);
  float s2 = 0.f;
  #pragma unroll
  for (int k = 0; k < 27; ++k) { float d = iou[k] - mean; s2 += d * d; }
  float thr = mean + sqrtf(s2 * (1.0f / 26.0f));   // ddof=1 -> /26

  #pragma unroll
  for (int k = 0; k < 27; ++k) {
    if (iou[k] > thr) {
      int a = idx[k];
      const float4 av = *(const float4*)(anchors + ((size_t)a << 2));
      float acx = (av.x + av.z) * 0.5f;
      float acy = (av.y + av.w) * 0.5f;
      float l_ = acx - gx1, t_ = acy - gy1, r_ = gx2 - acx, b_ = gy2 - acy;
      float mn = fminf(fminf(l_, t_), fminf(r_, b_));
      if (mn > 1e-9f) {                      // is_in_gts, EPS = 1e-9
        atomicAdd(&cnt[(size_t)b * L_ANCH + a], 1);
        atomicMin(&firstg[(size_t)b * L_ANCH + a], g);   // argmax of binary mask = first g
      }
    }
  }
}

// -------------------- K3: conflict resolution + labels + bboxes (per b,a)
__global__ __launch_bounds__(256) void k3_final(const float* __restrict__ anchors,
                                                const float* __restrict__ gt,
                                                const int* __restrict__ glabels,
                                                const int* __restrict__ bgp,
                                                const int* __restrict__ cnt,
                                                const int* __restrict__ firstg,
                                                float* __restrict__ out_labels,
                                                float* __restrict__ out_bbox) {
  __shared__ float4 sgt[N_GTS];
  __shared__ int    slab[N_GTS];
  int b   = blockIdx.y;
  int tid = threadIdx.x;
  for (int i = tid; i < N_GTS; i += 256) {
    sgt[i]  = *(const float4*)(gt + ((size_t)(b * N_GTS + i) << 2));
    slab[i] = glabels[b * N_GTS + i];
  }
  __syncthreads();

  int a = blockIdx.x * 256 + tid;
  if (a >= L_ANCH) return;
  size_t idx = (size_t)b * L_ANCH + a;

  int c = cnt[idx];
  int gsel;
  if (c == 0) {
    gsel = -1;
  } else if (c == 1) {
    gsel = firstg[idx];
  } else {
    // conflict: argmax over ALL 128 gts (incl. padded), first max kept
    const float4 av = *(const float4*)(anchors + ((size_t)a << 2));
    float aarea = (av.z - av.x) * (av.w - av.y);
    float best = -1.0f; int bi = 0;
    for (int g = 0; g < N_GTS; ++g) {
      float4 gv = sgt[g];
      float lx = fmaxf(gv.x, av.x), ly = fmaxf(gv.y, av.y);
      float rx = fminf(gv.z, av.z), ry = fminf(gv.w, av.w);
      float wx = rx - lx; wx = wx < 0.f ? 0.f : wx;
      float wy = ry - ly; wy = wy < 0.f ? 0.f : wy;
      float inter = wx * wy;
      float garea = (gv.z - gv.x) * (gv.w - gv.y);
      float v = inter / (garea + aarea - inter + 1e-9f);
      if (v > best) { best = v; bi = g; }
    }
    gsel = bi;
  }

  int bg  = bgp[0];
  int lab = (gsel >= 0) ? slab[gsel] : bg;
  int gi  = (gsel < 0) ? 0 : gsel;          // reference gathers gt 0 for background
  out_labels[idx] = (float)lab;
  ((float4*)out_bbox)[idx] = sgt[gi];
}

// --------------- K4: one-hot scores, 128-bit coalesced stores (86 MB write)
// NUM_CLASSES (80) % 4 == 0, so a float4 never crosses an (b,a) row boundary.
__global__ __launch_bounds__(256) void k4_scores(const float* __restrict__ labels,
                                                 float4* __restrict__ scores,
                                                 int total4) {
  int i = blockIdx.x * 256 + threadIdx.x;
  if (i >= total4) return;
  int ba = i / (NUM_CLASSES / 4);            // 20 float4 per (b,a) row
  int c0 = (i - ba * (NUM_CLASSES / 4)) * 4; // first class of this quad
  int lab = (int)labels[ba];                 // bg label (80) never matches
  float4 v;
  v.x = (lab == c0 + 0) ? 1.0f : 0.0f;
  v.y = (lab == c0 + 1) ? 1.0f : 0.0f;
  v.z = (lab == c0 + 2) ? 1.0f : 0.0f;
  v.w = (lab == c0 + 3) ? 1.0f : 0.0f;
  scores[i] = v;
}

// ---------------------------------------------------------------- launcher
extern "C" void kernel_launch(void* const* d_in, const int* in_sizes, int n_in,
                              void* d_out, int out_size, void* d_ws, size_t ws_size,
                              hipStream_t stream) {
  const float* anchors = (const float*)d_in[0];   // (L,4)
  const int*   glabels = (const int*)d_in[1];     // (B,128,1)
  const float* gt      = (const float*)d_in[2];   // (B,128,4)
  const float* pad     = (const float*)d_in[3];   // (B,128,1)
  const int*   bgp     = (const int*)d_in[4];     // scalar

  // workspace: topk idx (B*N*27 ints), claim count + first-gt (B*L ints each)
  char* ws = (char*)d_ws;
  int* topk   = (int*)ws;                               // 110,592 B (256-aligned)
  int* cnt    = (int*)(ws + 110592);                    // 1,075,200 B
  int* firstg = (int*)(ws + 110592 + 1075200);          // 1,075,200 B

  const int BL = B_SZ * L_ANCH;                         // 268800
  float* out_labels = (float*)d_out;                    // BL
  float* out_bbox   = out_labels + BL;                  // BL*4 (16B aligned)
  float* out_scores = out_bbox + (size_t)BL * 4;        // BL*80 (16B aligned)

  k0_init<<<(BL + 255) / 256, 256, 0, stream>>>(cnt, firstg, BL);
  k1_topk<<<64 * 3, 32, 0, stream>>>(anchors, gt, topk);
  k2_thresh<<<(B_SZ * N_GTS + 127) / 128, 128, 0, stream>>>(anchors, gt, pad, topk,
                                                            cnt, firstg);
  dim3 g3((L_ANCH + 255) / 256, B_SZ);
  k3_final<<<g3, 256, 0, stream>>>(anchors, gt, glabels, bgp, cnt, firstg,
                                   out_labels, out_bbox);
  int total4 = BL * (NUM_CLASSES / 4);                  // 5,376,000 float4 stores
  k4_scores<<<(total4 + 255) / 256, 256, 0, stream>>>(out_labels,
                                                      (float4*)out_scores, total4);
}